// yolo_detect_target_56453050139251
// MI455X (gfx1250) — compile-verified
//
#include <hip/hip_runtime.h>
#include <stdint.h>

#define N_CLASSES       80
#define CONF_TH         0.25f
#define TILE_ROWS       256
#define SUB_ROWS        64
#define NSUB            4
#define CHUNKS_PER_THR  5      // 64 rows * 320 B / 16 B / 256 threads
#define SUM_BLOCKS      512
#define SUM_THREADS     256

#if __has_builtin(__builtin_amdgcn_wmma_f32_16x16x4_f32)
#define HAVE_WMMA_F32X4 1
typedef float v2f_t __attribute__((ext_vector_type(2)));
typedef float v8f_t __attribute__((ext_vector_type(8)));
#endif

// Issue one 64-row (20 KB) sub-tile as 1280 coalesced async b128 copies into LDS.
// lds_off is the 32-bit LDS byte address of the destination buffer.
__device__ __forceinline__ void async_issue_subtile(const char* gbase, unsigned lds_off, int tid) {
#pragma unroll
  for (int i = 0; i < CHUNKS_PER_THR; ++i) {
    int idx = i * 256 + tid;                       // consecutive lanes -> consecutive 16B chunks
    unsigned long long g = (unsigned long long)(uintptr_t)gbase + (unsigned)(idx * 16);
    unsigned l = lds_off + (unsigned)(idx * 16);
    asm volatile("global_load_async_to_lds_b128 %0, %1, off"
                 :: "v"(l), "v"(g) : "memory");
  }
}

__device__ __forceinline__ void wait_async_all() {
  asm volatile("s_wait_asynccnt 0" ::: "memory");
}

// Phase 0: reset the first-fail cutoff each call (harness poisons ws only once).
__global__ void k_init(unsigned* cutoff, unsigned v) {
  if (threadIdx.x == 0 && blockIdx.x == 0) *cutoff = v;
}

// Phase 1: per-row max over 80 classes; record first row failing conf>=0.25.
// Block = 256 threads = one 256-row tile; double-buffered async LDS staging.
__global__ void __launch_bounds__(256) k_rowmax(const float* __restrict__ post,
                                                float* __restrict__ maxconf,
                                                unsigned* __restrict__ cutoff,
                                                int loop_count) {
  __shared__ float lds[2][SUB_ROWS * N_CLASSES];   // 2 * 20 KB
  const int tid = threadIdx.x;
  const int tile_base = blockIdx.x * TILE_ROWS;
  const char* gpost = (const char*)post;
  const unsigned lds_off0 = (unsigned)(uintptr_t)&lds[0][0];  // low 32 bits = LDS offset
  const unsigned lds_off1 = (unsigned)(uintptr_t)&lds[1][0];

  async_issue_subtile(gpost + (size_t)tile_base * (N_CLASSES * 4), lds_off0, tid);
  wait_async_all();
  __syncthreads();

  const int r4 = tid >> 2;        // row within sub-tile (0..63)
  const int lane4 = tid & 3;      // 4 lanes cooperate per row
  int buf = 0;
#pragma unroll
  for (int s = 0; s < NSUB; ++s) {
    if (s + 1 < NSUB) {
      async_issue_subtile(gpost + (size_t)(tile_base + (s + 1) * SUB_ROWS) * (N_CLASSES * 4),
                          buf ? lds_off0 : lds_off1, tid);
    }
    // Consume current buffer while the async engine fills the other one.
    const float4* rp = reinterpret_cast<const float4*>(&lds[buf][r4 * N_CLASSES]);
    float m = -3.4e38f;
#pragma unroll
    for (int i = 0; i < 5; ++i) {
      float4 v = rp[lane4 + 4 * i];                // ds_load_b128, conflict-free
      m = fmaxf(m, fmaxf(fmaxf(v.x, v.y), fmaxf(v.z, v.w)));
    }
    m = fmaxf(m, __shfl_xor(m, 1));
    m = fmaxf(m, __shfl_xor(m, 2));
    int grow = tile_base + s * SUB_ROWS + r4;
    if (lane4 == 0 && grow < loop_count) {
      if (maxconf) maxconf[grow] = m;
      if (m < CONF_TH) atomicMin(cutoff, (unsigned)grow);
    }
    wait_async_all();       // next buffer landed (this wave) ...
    __syncthreads();        // ... and for all waves; also WAR-protects the buffer we just read
    buf ^= 1;
  }
}

// Phase 2a: sum maxconf + box coords over rows below the cutoff (deterministic partials).
__global__ void __launch_bounds__(SUM_THREADS) k_sum(const float* __restrict__ maxconf,
                                                     const float4* __restrict__ boxes,
                                                     const unsigned* __restrict__ cutoff,
                                                     float* __restrict__ partials,
                                                     int loop_count) {
  unsigned cu = *cutoff;
  int cut = (int)(cu < (unsigned)loop_count ? cu : (unsigned)loop_count);
  float acc = 0.f;
  for (int r = blockIdx.x * SUM_THREADS + threadIdx.x; r < cut; r += gridDim.x * SUM_THREADS) {
    float4 b = boxes[r];
    acc += maxconf[r] + ((b.x + b.y) + (b.z + b.w));
  }
  for (int o = 16; o > 0; o >>= 1) acc += __shfl_down(acc, o);
  __shared__ float wsum[SUM_THREADS / 32];
  int wid = threadIdx.x >> 5, lane = threadIdx.x & 31;
  if (lane == 0) wsum[wid] = acc;
  __syncthreads();
  if (threadIdx.x == 0) {
    float t = 0.f;
    for (int i = 0; i < SUM_THREADS / 32; ++i) t += wsum[i];
    partials[blockIdx.x] = t;
  }
}

// Phase 2b fallback (workspace too small to hold maxconf): recompute the row max.
__global__ void __launch_bounds__(SUM_THREADS) k_sum_recompute(const float* __restrict__ post,
                                                               const float4* __restrict__ boxes,
                                                               const unsigned* __restrict__ cutoff,
                                                               float* __restrict__ partials,
                                                               int loop_count) {
  unsigned cu = *cutoff;
  int cut = (int)(cu < (unsigned)loop_count ? cu : (unsigned)loop_count);
  float acc = 0.f;
  for (int r = blockIdx.x * SUM_THREADS + threadIdx.x; r < cut; r += gridDim.x * SUM_THREADS) {
    const float4* rowp = (const float4*)(post + (size_t)r * N_CLASSES);
    float m = -3.4e38f;
#pragma unroll
    for (int i = 0; i < N_CLASSES / 4; ++i) {
      float4 v = rowp[i];
      m = fmaxf(m, fmaxf(fmaxf(v.x, v.y), fmaxf(v.z, v.w)));
    }
    float4 b = boxes[r];
    acc += m + ((b.x + b.y) + (b.z + b.w));
  }
  for (int o = 16; o > 0; o >>= 1) acc += __shfl_down(acc, o);
  __shared__ float wsum[SUM_THREADS / 32];
  int wid = threadIdx.x >> 5, lane = threadIdx.x & 31;
  if (lane == 0) wsum[wid] = acc;
  __syncthreads();
  if (threadIdx.x == 0) {
    float t = 0.f;
    for (int i = 0; i < SUM_THREADS / 32; ++i) t += wsum[i];
    partials[blockIdx.x] = t;
  }
}

// Phase 3: reduce 512 partials -> scalar. Final 32->1 stage via a ones-matrix
// V_WMMA_F32_16X16X4_F32 (D = ones(16x4) x B: every D row = column sums; sum(D)=16*sum(B)).
__global__ void __launch_bounds__(256) k_final(const float* __restrict__ partials,
                                               float* __restrict__ out) {
  __shared__ float red[256];
  int tid = threadIdx.x;
  red[tid] = partials[tid] + partials[tid + 256];
  __syncthreads();
  if (tid < 32) {                 // exactly wave 0: EXEC all ones for WMMA
    float x = 0.f;
#pragma unroll
    for (int i = 0; i < 8; ++i) x += red[tid * 8 + i];
#ifdef HAVE_WMMA_F32X4
    v2f_t a; a.x = 1.0f; a.y = 1.0f;   // A = ones 16x4
    v2f_t b; b.x = x;    b.y = 0.0f;   // B carries the 32 lane partials
    v8f_t c = {};
    c = __builtin_amdgcn_wmma_f32_16x16x4_f32(false, a, false, b,
                                              (short)0, c, false, false);
    float y = ((c[0] + c[1]) + (c[2] + c[3])) + ((c[4] + c[5]) + (c[6] + c[7]));
    for (int o = 16; o > 0; o >>= 1) y += __shfl_xor(y, o);
    if (tid == 0) out[0] = y * (1.0f / 16.0f);
#else
    for (int o = 16; o > 0; o >>= 1) x += __shfl_xor(x, o);
    if (tid == 0) out[0] = x;
#endif
  }
}

extern "C" void kernel_launch(void* const* d_in, const int* in_sizes, int n_in,
                              void* d_out, int out_size, void* d_ws, size_t ws_size,
                              hipStream_t stream) {
  const float*  post  = (const float*)d_in[0];
  const float4* boxes = (const float4*)d_in[1];
  float* out = (float*)d_out;

  long long n = (long long)in_sizes[0] / N_CLASSES;   // 524288
  int loop = (int)(n * 4 / 5);                        // int(n*0.8) = 419430

  char* ws = (char*)d_ws;
  unsigned* cutoff = (unsigned*)ws;
  float* maxconf = (float*)(ws + 64);
  size_t mc_bytes = (((size_t)loop * 4) + 63) & ~(size_t)63;
  size_t need = 64 + mc_bytes + SUM_BLOCKS * 4;
  bool big = ws_size >= need;
  float* partials = big ? (float*)(ws + 64 + mc_bytes) : (float*)(ws + 64);

  k_init<<<1, 1, 0, stream>>>(cutoff, (unsigned)loop);

  int tiles = (loop + TILE_ROWS - 1) / TILE_ROWS;     // reads past loop stay < N_ROWS: safe
  k_rowmax<<<tiles, 256, 0, stream>>>(post, big ? maxconf : nullptr, cutoff, loop);

  if (big) k_sum<<<SUM_BLOCKS, SUM_THREADS, 0, stream>>>(maxconf, boxes, cutoff, partials, loop);
  else     k_sum_recompute<<<SUM_BLOCKS, SUM_THREADS, 0, stream>>>(post, boxes, cutoff, partials, loop);

  k_final<<<1, 256, 0, stream>>>(partials, out);
}